// DeepTENEncoding_63986422776020
// MI455X (gfx1250) — compile-verified
//
#include <hip/hip_runtime.h>
#include <math.h>

// ---------------------------------------------------------------------------
// DeepTEN encoding, fused single-pass over x (256 MB), fp32 WMMA on gfx1250.
// Round 3: async global->LDS staging (ASYNCcnt) + L2 prefetch of next tile.
// ---------------------------------------------------------------------------

typedef float v2f __attribute__((ext_vector_type(2)));
typedef float v8f __attribute__((ext_vector_type(8)));

#if defined(__has_builtin)
#  if __has_builtin(__builtin_amdgcn_global_load_async_to_lds_b32) && \
      __has_builtin(__builtin_amdgcn_s_wait_asynccnt)
#    define USE_ASYNC_LDS 1
#  endif
#endif
#ifndef USE_ASYNC_LDS
#  define USE_ASYNC_LDS 0
#endif

#define Bc 32
#define Dc 128
#define Nc 16384           // H*W
#define Kc 32
#define NT 128             // n-positions per inner tile
#define TILES_PER_WG 16    // chunk = 2048 positions
#define WG_PER_B (Nc / (NT * TILES_PER_WG))   // 8
#define XS_STRIDE 130      // padded D stride (bank-conflict free)
#define AS_STRIDE 34       // padded K stride

// LDS float layout
#define OFF_XS   0
#define OFF_CS   (OFF_XS + NT * XS_STRIDE)       // 16640
#define OFF_AS   (OFF_CS + Kc * XS_STRIDE)       // 20800
#define OFF_XSQ  (OFF_AS + NT * AS_STRIDE)       // 25152
#define OFF_CSQ  (OFF_XSQ + NT)                  // 25280
#define OFF_SSCL (OFF_CSQ + Kc)                  // 25312
#define OFF_SP   (OFF_SSCL + Kc)                 // 25344
#define SMEM_FLOATS (OFF_SP + 8 * 32)            // 25600
#define SMEM_BYTES (SMEM_FLOATS * 4)             // 102400 B

__global__ __launch_bounds__(256)
void enc_zero_kernel(float* __restrict__ p, int n) {
    int i = blockIdx.x * 256 + threadIdx.x;
    if (i < n) p[i] = 0.0f;
}

__global__ __launch_bounds__(256)
void enc_main_kernel(const float* __restrict__ x,
                     const float* __restrict__ cw,
                     const float* __restrict__ scale,
                     float* __restrict__ accE,    // [B][K][D]
                     float* __restrict__ accS)    // [B][K]
{
    extern __shared__ float smem[];
    float* Xs   = smem + OFF_XS;
    float* Cs   = smem + OFF_CS;
    float* As   = smem + OFF_AS;
    float* xsq  = smem + OFF_XSQ;
    float* csq  = smem + OFF_CSQ;
    float* sscl = smem + OFF_SSCL;
    float* Sp   = smem + OFF_SP;

    const int tid   = threadIdx.x;
    const int wave  = tid >> 5;
    const int lane  = tid & 31;
    const int lhalf = lane >> 4;     // 0|1 : lane half selects K-pair / M+8
    const int l16   = lane & 15;

    const int b     = blockIdx.x / WG_PER_B;
    const int chunk = blockIdx.x % WG_PER_B;
    const int nbase = chunk * (NT * TILES_PER_WG);

    // ---- stage codewords (K x D) into LDS, padded ----
    for (int i = 0; i < (Kc * Dc) / 256; ++i) {
        int idx = tid + i * 256;
        int k = idx >> 7, d = idx & 127;
        Cs[k * XS_STRIDE + d] = cw[idx];
    }
    __syncthreads();
    if (tid < Kc) {
        float s = 0.0f;
        for (int d = 0; d < Dc; ++d) { float c = Cs[tid * XS_STRIDE + d]; s += c * c; }
        csq[tid]  = s;
        sscl[tid] = scale[tid];
    }

    // ---- persistent GEMM2 accumulators: 2 (k-tile, d-tile) tiles per wave ----
    v8f e0 = {};
    v8f e1 = {};
    const int kt2 = wave & 1;        // k-tile: 0|1
    const int dt0 = wave >> 1;       // d-tiles: dt0 and dt0+4
    const int dt1 = dt0 + 4;
    float sreg = 0.0f;               // per-(wave,lane) partial of S[k=lane]

    const float* xb = x + ((size_t)b * Dc) * (size_t)Nc + nbase;

    for (int t = 0; t < TILES_PER_WG; ++t) {
        __syncthreads();   // previous iteration's readers of Xs/As are done

        // ---- stage x tile: x[b][d][n0+n] -> Xs[n][d]  (transpose) ----
        const float* xt = xb + (size_t)t * NT;
#if USE_ASYNC_LDS
        // CDNA5 async copy: per-lane LDS dest does the transpose in-flight,
        // no VGPR round-trip, tracked on ASYNCcnt.
        #pragma unroll
        for (int i = 0; i < (NT * Dc) / 256; ++i) {   // 64 x b32 per thread
            int idx = tid + i * 256;
            int n = idx & (NT - 1);
            int d = idx >> 7;
            __builtin_amdgcn_global_load_async_to_lds_b32(
                (__attribute__((address_space(1))) int*)(uintptr_t)(xt + (size_t)d * Nc + n),
                (__attribute__((address_space(3))) int*)&Xs[n * XS_STRIDE + d],
                0, 0);
        }
#else
        for (int i = 0; i < (NT * Dc) / 256; ++i) {   // 64 coalesced dword loads
            int idx = tid + i * 256;
            int n = idx & (NT - 1);
            int d = idx >> 7;
            Xs[n * XS_STRIDE + d] = xt[(size_t)d * Nc + n];
        }
#endif

        // ---- prefetch next tile into the 192MB L2 while this tile computes ----
        if (t + 1 < TILES_PER_WG) {
            const float* xn = xb + (size_t)(t + 1) * NT;
            #pragma unroll
            for (int j = 0; j < 2; ++j) {
                int idx  = tid + j * 256;   // 0..511 -> 512 cachelines (128B)
                int d    = idx >> 2;        // 0..127
                int line = idx & 3;         // 4 lines per 512B row
                __builtin_prefetch(xn + (size_t)d * Nc + line * 32, 0, 1);
            }
        }

#if USE_ASYNC_LDS
        __builtin_amdgcn_s_wait_asynccnt(0);
#endif
        __syncthreads();

        // ---- ||x||^2 per row ----
        if (tid < NT) {
            float s = 0.0f;
            #pragma unroll 8
            for (int d = 0; d < Dc; ++d) { float v = Xs[tid * XS_STRIDE + d]; s += v * v; }
            xsq[tid] = s;
        }
        __syncthreads();

        // ---- GEMM1: logits = -scale[k] * (x2 - 2*(x . c_k) + c2[k]) ----
        {
            const int mrow = wave * 16;
            const float* arow = &Xs[(mrow + l16) * XS_STRIDE + 2 * lhalf];
            #pragma unroll
            for (int kt = 0; kt < 2; ++kt) {
                v8f acc = {};
                const float* brow = &Cs[(kt * 16 + l16) * XS_STRIDE + 2 * lhalf];
                #pragma unroll
                for (int d0 = 0; d0 < Dc; d0 += 4) {
                    v2f a  = *(const v2f*)(arow + d0);
                    v2f bb = *(const v2f*)(brow + d0);
                    acc = __builtin_amdgcn_wmma_f32_16x16x4_f32(
                        false, a, false, bb, (short)0, acc, false, false);
                }
                const int   k  = kt * 16 + l16;
                const float sk = sscl[k];
                const float ck = csq[k];
                #pragma unroll
                for (int r = 0; r < 8; ++r) {
                    int   nl   = mrow + r + 8 * lhalf;         // C/D row mapping
                    float dist = xsq[nl] - 2.0f * acc[r] + ck;
                    As[nl * AS_STRIDE + k] = -sk * dist;
                }
            }
        }
        __syncthreads();

        // ---- softmax over K=32 per row ----
        if (tid < NT) {
            float* row = &As[tid * AS_STRIDE];
            float m = row[0];
            #pragma unroll
            for (int k = 1; k < Kc; ++k) m = fmaxf(m, row[k]);
            float s = 0.0f;
            #pragma unroll
            for (int k = 0; k < Kc; ++k) { float e = __expf(row[k] - m); row[k] = e; s += e; }
            float inv = 1.0f / s;
            #pragma unroll
            for (int k = 0; k < Kc; ++k) row[k] *= inv;
        }
        __syncthreads();

        // ---- partial S[k]: wave w sums its 16 rows, k = lane ----
        {
            const float* p = &As[(wave * 16) * AS_STRIDE + lane];
            float s = 0.0f;
            #pragma unroll
            for (int i = 0; i < 16; ++i) s += p[i * AS_STRIDE];
            sreg += s;
        }

        // ---- GEMM2: E += A^T (K x NT) * X (NT x D), accumulate in VGPRs ----
        {
            #pragma unroll 4
            for (int nb = 0; nb < NT; nb += 4) {
                const int na = nb + 2 * lhalf;
                v2f a, b0, b1;
                a.x  = As[(na + 0) * AS_STRIDE + kt2 * 16 + l16];
                a.y  = As[(na + 1) * AS_STRIDE + kt2 * 16 + l16];
                b0.x = Xs[(na + 0) * XS_STRIDE + dt0 * 16 + l16];
                b0.y = Xs[(na + 1) * XS_STRIDE + dt0 * 16 + l16];
                b1.x = Xs[(na + 0) * XS_STRIDE + dt1 * 16 + l16];
                b1.y = Xs[(na + 1) * XS_STRIDE + dt1 * 16 + l16];
                e0 = __builtin_amdgcn_wmma_f32_16x16x4_f32(
                    false, a, false, b0, (short)0, e0, false, false);
                e1 = __builtin_amdgcn_wmma_f32_16x16x4_f32(
                    false, a, false, b1, (short)0, e1, false, false);
            }
        }
    }

    // ---- flush E partials (8 contributions per element across the grid) ----
    float* eb = accE + (size_t)b * Kc * Dc;
    #pragma unroll
    for (int r = 0; r < 8; ++r) {
        int k = kt2 * 16 + r + 8 * lhalf;      // C/D row mapping
        atomicAdd(&eb[k * Dc + dt0 * 16 + l16], e0[r]);
        atomicAdd(&eb[k * Dc + dt1 * 16 + l16], e1[r]);
    }

    // ---- reduce S partials across waves, flush ----
    Sp[wave * 32 + lane] = sreg;
    __syncthreads();
    if (tid < Kc) {
        float s = 0.0f;
        #pragma unroll
        for (int w = 0; w < 8; ++w) s += Sp[w * 32 + tid];
        atomicAdd(&accS[b * Kc + tid], s);
    }
}

__global__ __launch_bounds__(256)
void enc_finalize_kernel(const float* __restrict__ accE,
                         const float* __restrict__ accS,
                         const float* __restrict__ cw,
                         float* __restrict__ out)
{
    int i = blockIdx.x * 256 + threadIdx.x;   // i in [0, B*K*D)
    if (i < Bc * Kc * Dc) {
        int b = i >> 12;               // /(K*D)
        int k = (i >> 7) & (Kc - 1);
        int d = i & (Dc - 1);
        out[i] = accE[i] - accS[b * Kc + k] * cw[k * Dc + d];
    }
}

extern "C" void kernel_launch(void* const* d_in, const int* in_sizes, int n_in,
                              void* d_out, int out_size, void* d_ws, size_t ws_size,
                              hipStream_t stream) {
    const float* x     = (const float*)d_in[0];   // (B, D, H, W)
    const float* cw    = (const float*)d_in[1];   // (K, D)
    const float* scale = (const float*)d_in[2];   // (K,)
    float* out = (float*)d_out;                   // (B, K*D)

    float* accE = (float*)d_ws;                   // B*K*D floats
    float* accS = accE + Bc * Kc * Dc;            // B*K floats

    const int accTotal = Bc * Kc * Dc + Bc * Kc;  // 132096
    enc_zero_kernel<<<(accTotal + 255) / 256, 256, 0, stream>>>(accE, accTotal);

    enc_main_kernel<<<Bc * WG_PER_B, 256, SMEM_BYTES, stream>>>(
        x, cw, scale, accE, accS);

    enc_finalize_kernel<<<(Bc * Kc * Dc + 255) / 256, 256, 0, stream>>>(
        accE, accS, cw, out);
}